// BandPassFilter_39333310496998
// MI455X (gfx1250) — compile-verified
//
#include <hip/hip_runtime.h>

typedef __attribute__((ext_vector_type(2))) float v2f;
typedef __attribute__((ext_vector_type(8))) float v8f;

#define BATCH   64
#define LSEQ    8192
#define NBANDS  20
#define KTAP    769          // FIR length
#define PADLEN  2307         // min(3*K, L-1)
#define KFULL   1537         // 2*KTAP-1 autocorrelation taps
#define KPAD    1540         // padded to multiple of 4 (== 11*140)
#define MPAD    32           // 20 bands padded to 2 WMMA M-tiles
#define CK      140          // K chunk staged in LDS (multiple of 4)
#define TBLK    512          // output t per block (8 waves * 64)
#define NP      (TBLK + KPAD) // xp window floats per block = 2052

// Reflection constants:
//   padded xp index p in [0, 12806):
//     p < 2307            : xp[p] = -xs[2307 - p]
//     2307 <= p < 10499   : xp[p] =  xs[p - 2307]
//     p >= 10499          : xp[p] = -xs[8190 - (p - 10499)] = -xs[18689 - p]
#define PR_MID  (PADLEN + LSEQ)            // 10499
#define PR_RIDX (PR_MID + LSEQ - 2)        // 18689

// ---------------------------------------------------------------------------
// Kernel 1: per-band autocorrelation  c[o][k] = sum_i w[i]*w[i + (k-768)]
// Rows o>=20 and cols k>=1537 are zero (padding for the WMMA GEMM).
// ---------------------------------------------------------------------------
__global__ __launch_bounds__(256)
void bpf_autocorr_kernel(const float* __restrict__ kern, float* __restrict__ cpad) {
    __shared__ float w[KTAP];
    const int o   = blockIdx.x;     // 0..31
    const int tid = threadIdx.x;
    if (o < NBANDS) {
        for (int i = tid; i < KTAP; i += 256) w[i] = kern[o * KTAP + i];
    }
    __syncthreads();
    for (int k = tid; k < KPAD; k += 256) {
        float s = 0.0f;
        if (o < NBANDS && k < KFULL) {
            const int d  = k - (KTAP - 1);
            const int i0 = d < 0 ? -d : 0;
            const int i1 = d > 0 ? KTAP - d : KTAP;
            for (int i = i0; i < i1; ++i) s += w[i] * w[i + d];
        }
        cpad[o * KPAD + k] = s;
    }
}

// ---------------------------------------------------------------------------
// Kernel 2: banded GEMM via V_WMMA_F32_16X16X4_F32.
//   out[b,o,t] = sum_k cpad[o][k] * xp[b, t + 1539 + k]
// xp is the odd-reflection padded signal (built on the fly into LDS).
// Each block: one batch b, 512 t values. Each wave: 32(M) x 64(N) tile.
// ---------------------------------------------------------------------------
__global__ __launch_bounds__(256)
void bpf_filt_kernel(const float* __restrict__ x,
                     const float* __restrict__ cpad,
                     float* __restrict__ out) {
    __shared__ float xp[NP];           // 8208 B
    __shared__ float cl[MPAD * CK];    // 17920 B

    const int tid  = threadIdx.x;
    const int lane = tid & 31;
    const int wave = tid >> 5;
    const int b    = blockIdx.x >> 4;            // 16 t-tiles per batch
    const int tblk = (blockIdx.x & 15) * TBLK;

    const float* __restrict__ xs = x + b * LSEQ;

    // Stage reflected window: xp[i] = xp_global[tblk + 1539 + i]
    for (int i = tid; i < NP; i += 256) {
        const int p = tblk + (PADLEN - (KTAP - 1)) + i;   // tblk + 1539 + i
        float v;
        if (p < PADLEN)       v = -xs[PADLEN - p];
        else if (p < PR_MID)  v =  xs[p - PADLEN];
        else                  v = -xs[PR_RIDX - p];
        xp[i] = v;
    }

    v8f acc[2][4];
#pragma unroll
    for (int m = 0; m < 2; ++m)
#pragma unroll
        for (int j = 0; j < 4; ++j)
            acc[m][j] = (v8f){0.f, 0.f, 0.f, 0.f, 0.f, 0.f, 0.f, 0.f};

    const int woff = wave * 64;             // wave's N offset within block
    const int nrow = lane & 15;             // M row (A) / N col (B)
    const int koff = (lane < 16) ? 0 : 2;   // K sub-offset per half-wave

    for (int kc = 0; kc < KPAD; kc += CK) {
        __syncthreads();                    // protect cl reuse + xp on iter 0
        for (int idx = tid; idx < MPAD * CK; idx += 256) {
            const int row = idx / CK;
            const int ck  = idx - row * CK;
            cl[idx] = cpad[row * KPAD + kc + ck];
        }
        __syncthreads();

        for (int kk = 0; kk < CK; kk += 4) {
            // A fragments (16x4 f32): lane -> c[row=nrow][kk+koff .. +1]
            const float* arow = &cl[nrow * CK + kk + koff];
            v2f a0, a1;
            a0.x = arow[0];            a0.y = arow[1];
            a1.x = arow[16 * CK];      a1.y = arow[16 * CK + 1];
            const int kglob = kc + kk;
#pragma unroll
            for (int j = 0; j < 4; ++j) {
                // B fragment (4x16 f32): B[kk'][n] = xp[woff + j*16 + n + kglob + kk']
                const int bi = woff + j * 16 + nrow + kglob + koff;
                v2f bb;
                bb.x = xp[bi];
                bb.y = xp[bi + 1];
                acc[0][j] = __builtin_amdgcn_wmma_f32_16x16x4_f32(
                    false, a0, false, bb, (short)0, acc[0][j], false, false);
                acc[1][j] = __builtin_amdgcn_wmma_f32_16x16x4_f32(
                    false, a1, false, bb, (short)0, acc[1][j], false, false);
            }
        }
    }

    // Store: VGPR r holds M = r (lanes 0-15) or r+8 (lanes 16-31); N = lane%16
    float* __restrict__ ob = out + b * (NBANDS * LSEQ);
    const int orow  = (lane < 16) ? 0 : 8;
    const int tbase = tblk + woff + nrow;
#pragma unroll
    for (int j = 0; j < 4; ++j) {
        const int t = tbase + j * 16;
#pragma unroll
        for (int r = 0; r < 8; ++r) {
            const int o0 = r + orow;          // tile 0: o in [0,16)
            ob[o0 * LSEQ + t] = acc[0][j][r];
            const int o1 = 16 + r + orow;     // tile 1: only o<20 valid
            if (o1 < NBANDS) ob[o1 * LSEQ + t] = acc[1][j][r];
        }
    }
}

extern "C" void kernel_launch(void* const* d_in, const int* in_sizes, int n_in,
                              void* d_out, int out_size, void* d_ws, size_t ws_size,
                              hipStream_t stream) {
    const float* x    = (const float*)d_in[0];   // (64, 1, 8192) f32
    const float* kern = (const float*)d_in[1];   // (20, 769)     f32
    float* out  = (float*)d_out;                 // (64, 1, 20, 8192) f32
    float* cpad = (float*)d_ws;                  // 32*1540 f32 = 197120 B

    bpf_autocorr_kernel<<<MPAD, 256, 0, stream>>>(kern, cpad);
    bpf_filt_kernel<<<BATCH * (LSEQ / TBLK), 256, 0, stream>>>(x, cpad, out);
}